// unit_GAT_55602646614507
// MI455X (gfx1250) — compile-verified
//
#include <hip/hip_runtime.h>
#include <hip/hip_bf16.h>
#include <math.h>

#define N_NODES 100000
#define N_EDGES 1000000
#define DIM 64
#define NEG_SLOPE 0.2f

typedef __attribute__((ext_vector_type(2))) float v2f;
typedef __attribute__((ext_vector_type(4))) float v4f;
typedef __attribute__((ext_vector_type(8))) float v8f;

// Monotone float -> uint mapping so we can do float segment-max with u32 atomicMax.
__device__ __forceinline__ unsigned int fkey(float f) {
  unsigned int u = __float_as_uint(f);
  return (u & 0x80000000u) ? ~u : (u ^ 0x80000000u);
}
__device__ __forceinline__ float funkey(unsigned int k) {
  return (k & 0x80000000u) ? __uint_as_float(k ^ 0x80000000u)
                           : __uint_as_float(~k);
}

// ---------------------------------------------------------------------------
// K1: h = x @ W via V_WMMA_F32_16X16X4_F32. 8 waves/block, 16 rows per wave,
// W (64x64 f32, 16 KB) staged in LDS once per block.
// ---------------------------------------------------------------------------
__global__ __launch_bounds__(256) void k_gemm_wmma(const float* __restrict__ x,
                                                   const float* __restrict__ W,
                                                   float* __restrict__ h,
                                                   int nTiles) {
  __shared__ float Ws[DIM * DIM];
  {
    const v4f* Wv = (const v4f*)W;
    v4f* Wsv = (v4f*)Ws;
    int t = threadIdx.x;
#pragma unroll
    for (int i = 0; i < 4; ++i) Wsv[t + 256 * i] = Wv[t + 256 * i];
  }
  __syncthreads();

  const int wave = threadIdx.x >> 5;
  const int lane = threadIdx.x & 31;
  const int rowTile = blockIdx.x * 8 + wave;   // wave-uniform
  if (rowTile >= nTiles) return;               // EXEC stays all-ones per wave

  const int half = lane >> 4;   // 0: K pair (k0,k0+1) / rows r..r+7 ; 1: (k0+2,k0+3) / r+8..
  const int lid  = lane & 15;
  const int row  = rowTile * 16 + lid;
  const float* xr = x + (size_t)row * DIM;

  // speculative prefetch of a later tile's A data (harmless if OOB)
  __builtin_prefetch(xr + (size_t)128 * DIM, 0, 0);

  v8f acc0 = {}, acc1 = {}, acc2 = {}, acc3 = {};
#pragma unroll
  for (int k0 = 0; k0 < DIM; k0 += 4) {
    v2f a;
    a.x = xr[k0 + 2 * half];
    a.y = xr[k0 + 2 * half + 1];
    const float* w0 = Ws + (k0 + 2 * half) * DIM;
    const float* w1 = w0 + DIM;
    v2f b0, b1, b2, b3;
    b0.x = w0[lid +  0]; b0.y = w1[lid +  0];
    b1.x = w0[lid + 16]; b1.y = w1[lid + 16];
    b2.x = w0[lid + 32]; b2.y = w1[lid + 32];
    b3.x = w0[lid + 48]; b3.y = w1[lid + 48];
    acc0 = __builtin_amdgcn_wmma_f32_16x16x4_f32(false, a, false, b0, (short)0, acc0, false, false);
    acc1 = __builtin_amdgcn_wmma_f32_16x16x4_f32(false, a, false, b1, (short)0, acc1, false, false);
    acc2 = __builtin_amdgcn_wmma_f32_16x16x4_f32(false, a, false, b2, (short)0, acc2, false, false);
    acc3 = __builtin_amdgcn_wmma_f32_16x16x4_f32(false, a, false, b3, (short)0, acc3, false, false);
  }

  // C/D layout: VGPR r -> row (r + 8*half), col (tile*16 + lid)
  float* hb = h + ((size_t)rowTile * 16 + 8 * half) * DIM + lid;
#pragma unroll
  for (int r = 0; r < 8; ++r) {
    hb[(size_t)r * DIM +  0] = acc0[r];
    hb[(size_t)r * DIM + 16] = acc1[r];
    hb[(size_t)r * DIM + 32] = acc2[r];
    hb[(size_t)r * DIM + 48] = acc3[r];
  }
}

// ---------------------------------------------------------------------------
// K0: init per-node state: segment-max keys = 0 (== -NaN, below every real key),
// segment sums = 0.
// ---------------------------------------------------------------------------
__global__ void k_init_nodes(unsigned int* __restrict__ mkey,
                             float* __restrict__ s, int n) {
  int i = blockIdx.x * blockDim.x + threadIdx.x;
  if (i >= n) return;
  mkey[i] = 0u;
  s[i] = 0.0f;
}

// ---------------------------------------------------------------------------
// K2: el[i] = h[i,:].attn_l ; er[i] = h[i,:].attn_r   (h is L2-resident)
// ---------------------------------------------------------------------------
__global__ void k_node_scores(const float* __restrict__ h,
                              const float* __restrict__ al,
                              const float* __restrict__ ar,
                              float* __restrict__ el,
                              float* __restrict__ er, int n) {
  int i = blockIdx.x * blockDim.x + threadIdx.x;
  if (i >= n) return;
  const v4f* hr  = (const v4f*)(h + (size_t)i * DIM);
  const v4f* alv = (const v4f*)al;
  const v4f* arv = (const v4f*)ar;
  float sl = 0.f, sr = 0.f;
#pragma unroll
  for (int j = 0; j < DIM / 4; ++j) {
    v4f hv = hr[j], a = alv[j], b = arv[j];
    sl += hv.x * a.x + hv.y * a.y + hv.z * a.z + hv.w * a.w;
    sr += hv.x * b.x + hv.y * b.y + hv.z * b.z + hv.w * b.w;
  }
  el[i] = sl;
  er[i] = sr;
}

// ---------------------------------------------------------------------------
// K3: e = leaky_relu(el[src]+er[dst]); segment max via u32 atomicMax.
// ---------------------------------------------------------------------------
__global__ void k_edge_score_max(const float* __restrict__ el,
                                 const float* __restrict__ er,
                                 const int* __restrict__ src,
                                 const int* __restrict__ dst,
                                 float* __restrict__ e,
                                 unsigned int* __restrict__ mkey, int nE) {
  int i = blockIdx.x * blockDim.x + threadIdx.x;
  if (i >= nE) return;
  float v = el[src[i]] + er[dst[i]];
  v = v > 0.0f ? v : NEG_SLOPE * v;
  e[i] = v;
  atomicMax(&mkey[dst[i]], fkey(v));
}

// ---------------------------------------------------------------------------
// K4: ex = exp(e - m[dst]); segment sum via global_atomic_add_f32.
// ---------------------------------------------------------------------------
__global__ void k_edge_exp_sum(const int* __restrict__ dst,
                               const unsigned int* __restrict__ mkey,
                               float* __restrict__ e,
                               float* __restrict__ s, int nE) {
  int i = blockIdx.x * blockDim.x + threadIdx.x;
  if (i >= nE) return;
  int d = dst[i];
  float ex = __expf(e[i] - funkey(mkey[d]));
  e[i] = ex;
  atomicAdd(&s[d], ex);
}

// ---------------------------------------------------------------------------
// K5: out = x + bias (residual + bias; agg accumulates on top via atomics)
// ---------------------------------------------------------------------------
__global__ void k_init_out(const float* __restrict__ x,
                           const float* __restrict__ bias,
                           float* __restrict__ out, int total) {
  int i = blockIdx.x * blockDim.x + threadIdx.x;
  if (i >= total) return;
  out[i] = x[i] + bias[i & (DIM - 1)];
}

// ---------------------------------------------------------------------------
// K6: out[dst] += alpha * h[src]. 16 threads per edge, 4 cols each ->
// consecutive lanes hit consecutive addresses (coalesced gathers + atomics).
// ---------------------------------------------------------------------------
__global__ void k_edge_aggregate(const int* __restrict__ src,
                                 const int* __restrict__ dst,
                                 const float* __restrict__ ex,
                                 const float* __restrict__ s,
                                 const float* __restrict__ h,
                                 float* __restrict__ out, int nE) {
  int idx = blockIdx.x * blockDim.x + threadIdx.x;
  int edge = idx >> 4;
  if (edge >= nE) return;
  int c4 = (idx & 15) * 4;
  int sN = src[edge], dN = dst[edge];
  float alpha = ex[edge] / fmaxf(s[dN], 1e-9f);
  v4f hv = *(const v4f*)(h + (size_t)sN * DIM + c4);
  float* o = out + (size_t)dN * DIM + c4;
  atomicAdd(o + 0, alpha * hv.x);
  atomicAdd(o + 1, alpha * hv.y);
  atomicAdd(o + 2, alpha * hv.z);
  atomicAdd(o + 3, alpha * hv.w);
}

// ---------------------------------------------------------------------------
extern "C" void kernel_launch(void* const* d_in, const int* in_sizes, int n_in,
                              void* d_out, int out_size, void* d_ws, size_t ws_size,
                              hipStream_t stream) {
  const float* x     = (const float*)d_in[0];
  const float* W     = (const float*)d_in[1];
  const float* al    = (const float*)d_in[2];
  const float* ar    = (const float*)d_in[3];
  const float* bias  = (const float*)d_in[4];
  const int*   src   = (const int*)d_in[5];
  const int*   dst   = (const int*)d_in[6];
  float* out = (float*)d_out;

  char* ws = (char*)d_ws;
  size_t off = 0;
  float* h = (float*)(ws + off);          off += (size_t)N_NODES * DIM * sizeof(float);
  float* el = (float*)(ws + off);         off += (size_t)N_NODES * sizeof(float);
  float* er = (float*)(ws + off);         off += (size_t)N_NODES * sizeof(float);
  unsigned int* mkey = (unsigned int*)(ws + off); off += (size_t)N_NODES * sizeof(unsigned int);
  float* s = (float*)(ws + off);          off += (size_t)N_NODES * sizeof(float);
  float* e = (float*)(ws + off);          off += (size_t)N_EDGES * sizeof(float);
  (void)ws_size; (void)in_sizes; (void)n_in; (void)out_size;

  const int nTiles = N_NODES / 16;  // 6250, exact

  k_init_nodes<<<(N_NODES + 255) / 256, 256, 0, stream>>>(mkey, s, N_NODES);
  k_gemm_wmma<<<(nTiles + 7) / 8, 256, 0, stream>>>(x, W, h, nTiles);
  k_node_scores<<<(N_NODES + 255) / 256, 256, 0, stream>>>(h, al, ar, el, er, N_NODES);
  k_edge_score_max<<<(N_EDGES + 255) / 256, 256, 0, stream>>>(el, er, src, dst, e, mkey, N_EDGES);
  k_edge_exp_sum<<<(N_EDGES + 255) / 256, 256, 0, stream>>>(dst, mkey, e, s, N_EDGES);
  k_init_out<<<(N_NODES * DIM + 255) / 256, 256, 0, stream>>>(x, bias, out, N_NODES * DIM);
  k_edge_aggregate<<<((N_EDGES * 16) + 255) / 256, 256, 0, stream>>>(src, dst, e, s, h, out, N_EDGES);
}